// ProposalLayer_65704409694744
// MI455X (gfx1250) — compile-verified
//
#include <hip/hip_runtime.h>
#include <math.h>

// ---------------------------------------------------------------------------
// Faster-RCNN ProposalLayer for MI455X (gfx1250, wave32).
//   scores      : (1, 18, 512, 512) f32   d_in[0]
//   bbox_deltas : (1, 36, 512, 512) f32   d_in[1]
//   im_info     : (1, 3)            f32   d_in[2]
//   d_out       : rois (2000x5) ++ scores (2000x1) = 12000 f32
//
// Streaming passes run in channel-major (a, sp) order for fully coalesced
// HBM traffic (47 MB in + 9.4 MB keys out ~= 2.4 us at 23.3 TB/s); the
// reference (h,w,a) index is reconstructed only for ordering/tie-breaks.
// Pairwise NMS computes the IoU denominator area_i + area_j as a rank-2
// outer product on the matrix core (V_WMMA_F32_16X16X4_F32) per 16x32 tile.
// ---------------------------------------------------------------------------

#define N_ITEMS   2359296u          // 512*512*9
#define HW_FEAT   262144u           // 512*512
#define PRE_TOPN  12000
#define POST_TOPN 2000
#define CAND_CAP  16384
#define WORDS     375               // 12000/32
#define NMS_T     0.7f

typedef __attribute__((ext_vector_type(2))) float v2f;
typedef __attribute__((ext_vector_type(8))) float v8f;
typedef unsigned long long u64;

// Anchor widths/heights for base=16, ratios {0.5,1,2} x scales {8,16,32}.
// Centers are all (8 + 16*w, 8 + 16*h) under the reference's decode convention.
__constant__ float c_aw[9] = {184.f,368.f,736.f,128.f,256.f,512.f, 88.f,176.f,352.f};
__constant__ float c_ah[9] = { 96.f,192.f,384.f,128.f,256.f,512.f,176.f,352.f,704.f};

__device__ __forceinline__ unsigned f2key(float f) {
  unsigned u = __float_as_uint(f);
  return (u & 0x80000000u) ? ~u : (u | 0x80000000u);   // order-preserving
}

// Decode one item given (anchor a, spatial sp). sp = h*512 + w.
__device__ __forceinline__ void decode_asp(unsigned a, unsigned sp,
                                           const float* __restrict__ scores,
                                           const float* __restrict__ deltas,
                                           float im_h, float im_w, float ms,
                                           float bx[4], float* scm, int* valid) {
  unsigned wv = sp & 511u;
  unsigned hv = sp >> 9;
  float sc = scores[(size_t)(9u + a) * HW_FEAT + sp];
  const float* dp = deltas + (size_t)(4u * a) * HW_FEAT + sp;
  float d0 = dp[0];
  float d1 = dp[HW_FEAT];
  float d2 = dp[2u * HW_FEAT];
  float d3 = dp[3u * HW_FEAT];
  float aw = c_aw[a], ah = c_ah[a];
  float acx = 16.f * (float)wv + 8.f;
  float acy = 16.f * (float)hv + 8.f;
  float pcx = d0 * aw + acx;
  float pcy = d1 * ah + acy;
  float pw  = expf(d2) * aw;
  float ph  = expf(d3) * ah;
  float x1 = pcx - 0.5f * pw, x2 = pcx + 0.5f * pw;
  float y1 = pcy - 0.5f * ph, y2 = pcy + 0.5f * ph;
  x1 = fminf(fmaxf(x1, 0.f), im_w - 1.f);
  x2 = fminf(fmaxf(x2, 0.f), im_w - 1.f);
  y1 = fminf(fmaxf(y1, 0.f), im_h - 1.f);
  y2 = fminf(fmaxf(y2, 0.f), im_h - 1.f);
  int v = (x2 - x1 + 1.f >= ms) && (y2 - y1 + 1.f >= ms);
  bx[0] = x1; bx[1] = y1; bx[2] = x2; bx[3] = y2;
  *valid = v;
  *scm = v ? sc : -INFINITY;
}

// ------------------------------ utility ------------------------------------
__global__ void k_zero_u32(unsigned* p, unsigned n) {
  for (unsigned i = blockIdx.x * blockDim.x + threadIdx.x; i < n;
       i += gridDim.x * blockDim.x)
    p[i] = 0u;
}

// ---------------- pass 1: decode -> sortable key + hi histogram ------------
// Thread t -> (a = t/HW, sp = t%HW): all global accesses stride-1 per wave;
// anchor constants and channel bases become wave-uniform (SALU).
__global__ __launch_bounds__(256)
void k_decode_keys(const float* __restrict__ scores,
                   const float* __restrict__ deltas,
                   const float* __restrict__ im_info,
                   unsigned* __restrict__ keys,      // (a, sp) layout
                   unsigned* __restrict__ histHi) {
  unsigned t = blockIdx.x * blockDim.x + threadIdx.x;
  if (t >= N_ITEMS) return;
  unsigned a  = t >> 18;            // t / 262144
  unsigned sp = t & (HW_FEAT - 1u);
  float im_h = im_info[0], im_w = im_info[1], ms = 16.f * im_info[2];
  float bx[4], scm; int valid;
  decode_asp(a, sp, scores, deltas, im_h, im_w, ms, bx, &scm, &valid);
  unsigned key = f2key(scm);
  keys[t] = key;                    // fully coalesced store
  atomicAdd(&histHi[key >> 16], 1u);
}

// ---------------- radix-select scans (single block, 1024 thr) ---------------
// mode 0: find hi bucket B with count(key_hi > B) < 12000 <= +hist[B]
// mode 1: same within bucket B on low 16 bits -> threshold key T
__global__ __launch_bounds__(1024)
void k_scan(const unsigned* __restrict__ hist, unsigned* counters, int mode) {
  __shared__ unsigned sums[1024];
  int tid = threadIdx.x;
  int base = tid * 64;
  unsigned s = 0;
  for (int b = 0; b < 64; b++) s += hist[base + b];
  sums[tid] = s;
  __syncthreads();
  if (tid == 0) {                       // suffix-exclusive over chunks
    unsigned run = 0;
    for (int t = 1023; t >= 0; t--) { unsigned v = sums[t]; sums[t] = run; run += v; }
  }
  __syncthreads();
  unsigned target = (mode == 0) ? (unsigned)PRE_TOPN
                                : ((unsigned)PRE_TOPN - counters[2]);
  unsigned running = sums[tid];
  for (int b = 63; b >= 0; b--) {
    unsigned h = hist[base + b];
    if (running < target && running + h >= target) {
      if (mode == 0) { counters[1] = (unsigned)(base + b); counters[2] = running; }
      else           { counters[3] = (counters[1] << 16) | (unsigned)(base + b); }
    }
    running += h;
  }
}

__global__ __launch_bounds__(256)
void k_hist_lo(const unsigned* __restrict__ keys,
               const unsigned* __restrict__ counters,
               unsigned* __restrict__ histLo) {
  unsigned t = blockIdx.x * blockDim.x + threadIdx.x;
  if (t >= N_ITEMS) return;
  unsigned key = keys[t];
  if ((key >> 16) == counters[1]) atomicAdd(&histLo[key & 0xFFFFu], 1u);
}

// ---------------- compaction + padding + bitonic sort ----------------------
__global__ __launch_bounds__(256)
void k_compact(const unsigned* __restrict__ keys, unsigned* counters,
               u64* __restrict__ cand) {
  unsigned t = blockIdx.x * blockDim.x + threadIdx.x;
  if (t >= N_ITEMS) return;
  unsigned key = keys[t];
  if (key >= counters[3]) {
    unsigned a  = t >> 18;
    unsigned sp = t & (HW_FEAT - 1u);
    unsigned idx = sp * 9u + a;        // reference flat index (h,w,a order)
    unsigned slot = atomicAdd(&counters[0], 1u);
    if (slot < (unsigned)CAND_CAP)
      cand[slot] = ((u64)key << 32) | (u64)(0xFFFFFFFFu - idx);  // key desc, idx asc
  }
}

__global__ void k_pad(u64* cand, const unsigned* counters) {
  unsigned t = blockIdx.x * blockDim.x + threadIdx.x;
  if (t < (unsigned)CAND_CAP && t >= counters[0]) cand[t] = 0ull;
}

__global__ void k_bitonic(u64* cand, int kk, int jj) {
  int i = blockIdx.x * blockDim.x + threadIdx.x;
  int ixj = i ^ jj;
  if (ixj > i) {
    u64 a = cand[i], b = cand[ixj];
    bool descRegion = ((i & kk) == 0);
    if (descRegion ? (a < b) : (a > b)) { cand[i] = b; cand[ixj] = a; }
  }
}

// ---------------- gather top-12000 boxes/areas/scores ----------------------
__global__ __launch_bounds__(256)
void k_gather(const u64* __restrict__ cand,
              const float* __restrict__ scores,
              const float* __restrict__ deltas,
              const float* __restrict__ im_info,
              float* __restrict__ boxes, float* __restrict__ areas,
              float* __restrict__ sc12k, unsigned* __restrict__ validArr) {
  unsigned r = blockIdx.x * blockDim.x + threadIdx.x;
  if (r >= (unsigned)PRE_TOPN) return;
  u64 comp = cand[r];
  unsigned key = (unsigned)(comp >> 32);
  unsigned idx = 0xFFFFFFFFu - (unsigned)(comp & 0xFFFFFFFFull);
  if (key == 0u || idx >= N_ITEMS) {   // padding (shouldn't occur)
    boxes[r*4+0]=boxes[r*4+1]=boxes[r*4+2]=boxes[r*4+3]=0.f;
    areas[r]=1.f; sc12k[r]=0.f; validArr[r]=0u; return;
  }
  unsigned a  = idx % 9u;              // reference index -> (a, sp)
  unsigned sp = idx / 9u;
  float im_h = im_info[0], im_w = im_info[1], ms = 16.f * im_info[2];
  float bx[4], scm; int valid;
  decode_asp(a, sp, scores, deltas, im_h, im_w, ms, bx, &scm, &valid);
  boxes[r*4+0]=bx[0]; boxes[r*4+1]=bx[1]; boxes[r*4+2]=bx[2]; boxes[r*4+3]=bx[3];
  areas[r] = (bx[2]-bx[0]+1.f) * (bx[3]-bx[1]+1.f);
  sc12k[r] = scm;
  validArr[r] = (unsigned)valid;
}

__global__ void k_initkeep(const unsigned* __restrict__ validArr,
                           unsigned* __restrict__ keepw) {
  unsigned t = blockIdx.x * blockDim.x + threadIdx.x;
  if (t >= (unsigned)WORDS) return;
  unsigned w = 0;
  for (int b = 0; b < 32; b++) w |= (validArr[t * 32 + b] ? 1u : 0u) << b;
  keepw[t] = w;
}

// ---------------- suppression mask: one wave per 16x32 tile ----------------
// The IoU denominator area_i + area_j is a rank-2 outer product computed with
// V_WMMA_F32_16X16X4_F32 (A: [area_i, 1, 0, 0] rows; B: rows [1..1],[area_j],0,0).
__global__ __launch_bounds__(32)
void k_supmask(const float* __restrict__ boxes,
               const float* __restrict__ areas,
               unsigned* __restrict__ supmask) {
  int tj = blockIdx.x;             // 0..374  word (32 cols)
  int ti = blockIdx.y;             // 0..749  row tile (16 rows)
  int lane = threadIdx.x;          // 0..31, wave32, EXEC all-ones at WMMA
  int i0 = ti * 16, j0 = tj * 32;

  __shared__ float sib[16 * 4], sjb[32 * 4];
  __shared__ float sia[16], sja[32];
  {
    int j = j0 + lane;             // 12000 = 375*32, always in range
    sjb[lane*4+0] = boxes[j*4+0]; sjb[lane*4+1] = boxes[j*4+1];
    sjb[lane*4+2] = boxes[j*4+2]; sjb[lane*4+3] = boxes[j*4+3];
    sja[lane] = areas[j];
  }
  if (lane < 16) {
    int i = i0 + lane;             // 12000 = 750*16, always in range
    sib[lane*4+0] = boxes[i*4+0]; sib[lane*4+1] = boxes[i*4+1];
    sib[lane*4+2] = boxes[i*4+2]; sib[lane*4+3] = boxes[i*4+3];
    sia[lane] = areas[i];
  }
  __syncthreads();

  // A fragment (16x4 f32): lanes 0-15 hold K0,K1 of row m=lane; lanes 16-31 K2,K3.
  v2f a;
  a.x = (lane < 16) ? sia[lane] : 0.f;   // K0 = area_i
  a.y = (lane < 16) ? 1.f       : 0.f;   // K1 = 1
  unsigned bal[2][8];

  for (int t = 0; t < 2; t++) {
    // B fragment (4x16 f32), mirrored layout: lanes 0-15 hold K0,K1 of col n=lane.
    v2f b;
    b.x = (lane < 16) ? 1.f : 0.f;                 // B[0][n] = 1
    b.y = (lane < 16) ? sja[t * 16 + lane] : 0.f;  // B[1][n] = area_j
    v8f c = {};
    c = __builtin_amdgcn_wmma_f32_16x16x4_f32(false, a, false, b,
                                              (short)0, c, false, false);
    int n  = lane & 15;
    int jg = j0 + t * 16 + n;
    float jx1 = sjb[(t*16+n)*4+0], jy1 = sjb[(t*16+n)*4+1];
    float jx2 = sjb[(t*16+n)*4+2], jy2 = sjb[(t*16+n)*4+3];
    #pragma unroll
    for (int v = 0; v < 8; v++) {
      int m  = v + ((lane >= 16) ? 8 : 0);
      int ig = i0 + m;
      float ix1 = sib[m*4+0], iy1 = sib[m*4+1];
      float ix2 = sib[m*4+2], iy2 = sib[m*4+3];
      float xx1 = fmaxf(ix1, jx1), yy1 = fmaxf(iy1, jy1);
      float xx2 = fminf(ix2, jx2), yy2 = fminf(iy2, jy2);
      float iw = fmaxf(xx2 - xx1 + 1.f, 0.f);
      float ih = fmaxf(yy2 - yy1 + 1.f, 0.f);
      float inter = iw * ih;
      float uni = c[v] - inter;                 // area_i + area_j - inter (WMMA)
      bool pred = (inter > NMS_T * uni) && (jg > ig);
      bal[t][v] = __builtin_amdgcn_ballot_w32(pred);   // uniform across wave
    }
  }
  if (lane < 16) {
    unsigned w;
    if (lane < 8)
      w = (bal[0][lane] & 0xFFFFu) | ((bal[1][lane] & 0xFFFFu) << 16);
    else
      w = (bal[0][lane-8] >> 16) | (((bal[1][lane-8] >> 16) & 0xFFFFu) << 16);
    supmask[(size_t)(i0 + lane) * WORDS + tj] = w;
  }
}

// ---------------- sequential greedy NMS scan (single block) ----------------
__global__ __launch_bounds__(384)
void k_nms(const unsigned* __restrict__ sup, unsigned* keepw) {
  __shared__ unsigned kw[WORDS];
  int tid = threadIdx.x;
  if (tid < WORDS) kw[tid] = keepw[tid];
  __syncthreads();
  for (int i = 0; i < PRE_TOPN; i++) {
    unsigned bit = (kw[i >> 5] >> (i & 31)) & 1u;   // uniform read
    __syncthreads();
    if (bit) {
      if (tid < WORDS) kw[tid] &= ~sup[(size_t)i * WORDS + tid];
    }
    __syncthreads();
  }
  if (tid < WORDS) keepw[tid] = kw[tid];
}

// ---------------- rank + scatter into output -------------------------------
__global__ __launch_bounds__(384)
void k_finalize(const unsigned* __restrict__ keepw,
                const float* __restrict__ boxes,
                const float* __restrict__ sc12k,
                float* __restrict__ out) {
  __shared__ unsigned kw[WORDS];
  __shared__ unsigned base[WORDS];
  int tid = threadIdx.x;
  if (tid < WORDS) kw[tid] = keepw[tid];
  __syncthreads();
  if (tid == 0) {
    unsigned run = 0;
    for (int w = 0; w < WORDS; w++) { base[w] = run; run += __popc(kw[w]); }
  }
  __syncthreads();
  for (int i = tid; i < PRE_TOPN; i += 384) {
    int w = i >> 5, b = i & 31;
    if ((kw[w] >> b) & 1u) {
      unsigned rank = base[w] + __popc(kw[w] & ((1u << b) - 1u));
      if (rank < (unsigned)POST_TOPN) {
        out[rank*5+0] = 0.f;
        out[rank*5+1] = boxes[i*4+0];
        out[rank*5+2] = boxes[i*4+1];
        out[rank*5+3] = boxes[i*4+2];
        out[rank*5+4] = boxes[i*4+3];
        out[POST_TOPN*5 + rank] = sc12k[i];
      }
    }
  }
}

// ---------------------------------------------------------------------------
extern "C" void kernel_launch(void* const* d_in, const int* in_sizes, int n_in,
                              void* d_out, int out_size, void* d_ws, size_t ws_size,
                              hipStream_t stream) {
  const float* scores  = (const float*)d_in[0];
  const float* deltas  = (const float*)d_in[1];
  const float* im_info = (const float*)d_in[2];
  float* out = (float*)d_out;
  char* ws = (char*)d_ws;

  auto al = [](size_t x) { return (x + 255) & ~(size_t)255; };
  size_t off = 0;
  unsigned* keys    = (unsigned*)(ws + off); off = al(off + (size_t)N_ITEMS * 4);
  unsigned* histHi  = (unsigned*)(ws + off); off = al(off + 65536 * 4);
  unsigned* histLo  = (unsigned*)(ws + off); off = al(off + 65536 * 4);
  unsigned* counters= (unsigned*)(ws + off); off = al(off + 64 * 4);
  u64*      cand    = (u64*)     (ws + off); off = al(off + (size_t)CAND_CAP * 8);
  float*    boxes   = (float*)   (ws + off); off = al(off + (size_t)PRE_TOPN * 4 * 4);
  float*    areas   = (float*)   (ws + off); off = al(off + (size_t)PRE_TOPN * 4);
  float*    sc12k   = (float*)   (ws + off); off = al(off + (size_t)PRE_TOPN * 4);
  unsigned* validArr= (unsigned*)(ws + off); off = al(off + (size_t)PRE_TOPN * 4);
  unsigned* keepw   = (unsigned*)(ws + off); off = al(off + (size_t)WORDS * 4);
  unsigned* supmask = (unsigned*)(ws + off); off = al(off + (size_t)PRE_TOPN * WORDS * 4);
  (void)ws_size; (void)in_sizes; (void)n_in; (void)out_size;

  // zero: histograms, counters, output (unfilled slots must be 0)
  k_zero_u32<<<256, 256, 0, stream>>>(histHi, 65536u);
  k_zero_u32<<<256, 256, 0, stream>>>(histLo, 65536u);
  k_zero_u32<<<1, 64, 0, stream>>>(counters, 64u);
  k_zero_u32<<<48, 256, 0, stream>>>((unsigned*)out, 12000u);

  // decode + key + hi histogram (single coalesced streaming pass over 47 MB)
  k_decode_keys<<<N_ITEMS / 256, 256, 0, stream>>>(scores, deltas, im_info, keys, histHi);
  k_scan<<<1, 1024, 0, stream>>>(histHi, counters, 0);
  k_hist_lo<<<N_ITEMS / 256, 256, 0, stream>>>(keys, counters, histLo);
  k_scan<<<1, 1024, 0, stream>>>(histLo, counters, 1);

  // compact candidates >= threshold key, pad, bitonic sort descending
  k_compact<<<N_ITEMS / 256, 256, 0, stream>>>(keys, counters, cand);
  k_pad<<<CAND_CAP / 256, 256, 0, stream>>>(cand, counters);
  for (int k = 2; k <= CAND_CAP; k <<= 1)
    for (int j = k >> 1; j > 0; j >>= 1)
      k_bitonic<<<CAND_CAP / 256, 256, 0, stream>>>(cand, k, j);

  // gather top-12000, init keep mask from validity
  k_gather<<<(PRE_TOPN + 255) / 256, 256, 0, stream>>>(cand, scores, deltas, im_info,
                                                       boxes, areas, sc12k, validArr);
  k_initkeep<<<2, 256, 0, stream>>>(validArr, keepw);

  // pairwise suppression bitmask: 750x375 wave-tiles, WMMA area-sum per tile
  k_supmask<<<dim3(WORDS, PRE_TOPN / 16), 32, 0, stream>>>(boxes, areas, supmask);

  // sequential greedy scan + final scatter
  k_nms<<<1, 384, 0, stream>>>(supmask, keepw);
  k_finalize<<<1, 384, 0, stream>>>(keepw, boxes, sc12k, out);
}